// CH_D_65635690217699
// MI455X (gfx1250) — compile-verified
//
#include <hip/hip_runtime.h>
#include <hip/hip_bf16.h>
#include <math.h>

// ---------------------------------------------------------------------------
// CDNA5 (gfx1250) implementation of the depth/pose scoring network.
// conv2/conv3/conv4 run as implicit-GEMM on V_WMMA_F32_16X16X4_F32 (fp32).
// LDS input-tile staging uses GLOBAL_LOAD_ASYNC_TO_LDS_B32 (ASYNCcnt path)
// for interior tiles; boundary tiles fall back to a guarded scalar path.
// GroupNorm is computed with deterministic two-stage reductions and folded
// into per-channel affine tables applied fused with leaky-ReLU.
// ---------------------------------------------------------------------------

typedef float v2f __attribute__((ext_vector_type(2)));
typedef float v8f __attribute__((ext_vector_type(8)));

#define LRELU(t) ((t) < 0.f ? 0.2f * (t) : (t))

// ---------------- reductions (deterministic, no float atomics) --------------

__global__ void depth_partials(const float* __restrict__ d, int S, float* __restrict__ part) {
    __shared__ float ls[256];
    float s = 0.f;
    for (int i = blockIdx.x * 256 + threadIdx.x; i < S; i += gridDim.x * 256) s += d[i];
    ls[threadIdx.x] = s;
    __syncthreads();
    for (int w = 128; w > 0; w >>= 1) {
        if (threadIdx.x < w) ls[threadIdx.x] += ls[threadIdx.x + w];
        __syncthreads();
    }
    if (threadIdx.x == 0) part[blockIdx.x] = ls[0];
}

__global__ void finalize_mean(const float* __restrict__ part, int nb, float invS,
                              float* __restrict__ meanOut) {
    if (threadIdx.x == 0) {
        float s = 0.f;
        for (int b = 0; b < nb; ++b) s += part[b];
        *meanOut = s * invS;
    }
}

// per-channel partial sum/sumsq: grid = (NBLK, C), block = 256
__global__ void chan_partials(const float* __restrict__ src, int S,
                              float* __restrict__ psum, float* __restrict__ psq) {
    __shared__ float ls[256], lq[256];
    const int c = blockIdx.y;
    const float* p = src + (size_t)c * S;
    float s = 0.f, q = 0.f;
    for (int i = blockIdx.x * 256 + threadIdx.x; i < S; i += gridDim.x * 256) {
        float v = p[i];
        s += v;
        q += v * v;
    }
    ls[threadIdx.x] = s;
    lq[threadIdx.x] = q;
    __syncthreads();
    for (int w = 128; w > 0; w >>= 1) {
        if (threadIdx.x < w) {
            ls[threadIdx.x] += ls[threadIdx.x + w];
            lq[threadIdx.x] += lq[threadIdx.x + w];
        }
        __syncthreads();
    }
    if (threadIdx.x == 0) {
        psum[c * gridDim.x + blockIdx.x] = ls[0];
        psq[c * gridDim.x + blockIdx.x] = lq[0];
    }
}

// one block of C threads; folds GroupNorm into per-channel affine (A,B)
__global__ void finalize_gn(const float* __restrict__ psum, const float* __restrict__ psq,
                            int nblk, int C, int cpg, float S,
                            const float* __restrict__ gamma, const float* __restrict__ beta,
                            float* __restrict__ aff) {
    __shared__ float cs[64], cq[64];
    const int c = threadIdx.x;
    if (c < C) {
        float s = 0.f, q = 0.f;
        for (int b = 0; b < nblk; ++b) {
            s += psum[c * nblk + b];
            q += psq[c * nblk + b];
        }
        cs[c] = s;
        cq[c] = q;
    }
    __syncthreads();
    if (c < C) {
        const int g0 = (c / cpg) * cpg;
        float s = 0.f, q = 0.f;
        for (int k = 0; k < cpg; ++k) {
            s += cs[g0 + k];
            q += cq[g0 + k];
        }
        const float cnt = S * (float)cpg;
        const float mu = s / cnt;
        const float var = q / cnt - mu * mu;
        const float rs = rsqrtf(var + 1e-5f);
        const float A = rs * gamma[c];
        aff[2 * c] = A;
        aff[2 * c + 1] = beta[c] - mu * A;
    }
}

__global__ void apply_gn_lrelu(const float* __restrict__ src, const float* __restrict__ aff,
                               float* __restrict__ dst, int S) {
    const int c = blockIdx.y;
    const int i = blockIdx.x * 256 + threadIdx.x;
    if (i < S) {
        const float A = aff[2 * c], B = aff[2 * c + 1];
        const float t = src[(size_t)c * S + i] * A + B;
        dst[(size_t)c * S + i] = LRELU(t);
    }
}

// ---------------- input assembly -------------------------------------------

__global__ void build_x(const float* __restrict__ depth, const float* __restrict__ mask,
                        const float* __restrict__ meanP, float* __restrict__ x, int S) {
    const int i = blockIdx.x * 256 + threadIdx.x;
    if (i < S) {
        const float mu = *meanP;
        x[i] = (depth[i] - mu) * 10.f;
        x[S + i] = mask[i];
        x[2 * (size_t)S + i] = 0.f;
    }
}

__global__ void scatter_attn(const int* __restrict__ pairs, float* __restrict__ x2, int P) {
    const int p = blockIdx.x * 256 + threadIdx.x;
    if (p < P) x2[pairs[2 * p]] = 1.f;
}

// ---------------- conv1 (3 -> 32) direct (1.2% of FLOPs) --------------------

__global__ void conv1_direct(const float* __restrict__ x, const float* __restrict__ w,
                             const float* __restrict__ b, float* __restrict__ out,
                             int H_, int W_) {
    const int o = blockIdx.y;
    const int pix = blockIdx.x * 256 + threadIdx.x;
    if (pix >= H_ * W_) return;
    const int yy = pix / W_, xx = pix - yy * W_;
    float acc = b[o];
#pragma unroll
    for (int c = 0; c < 3; ++c)
#pragma unroll
        for (int ky = 0; ky < 3; ++ky) {
            const int gy = yy + ky - 1;
#pragma unroll
            for (int kx = 0; kx < 3; ++kx) {
                const int gx = xx + kx - 1;
                const float v = ((unsigned)gy < (unsigned)H_ && (unsigned)gx < (unsigned)W_)
                                    ? x[(size_t)c * H_ * W_ + (size_t)gy * W_ + gx]
                                    : 0.f;
                acc += v * w[((o * 3 + c) * 3 + ky) * 3 + kx];
            }
        }
    out[(size_t)o * H_ * W_ + pix] = acc;
}

// ---------------- weight swizzle for WMMA: dst[pos][cb][m][j] ---------------
// k-index ordering inside the GEMM: k = pos*CIN + cin, cin = cb*4 + j.
__global__ void prep_weights(const float* __restrict__ src, float* __restrict__ dst, int CIN) {
    const int t = blockIdx.x * blockDim.x + threadIdx.x;
    const int total = 64 * 9 * CIN;
    if (t >= total) return;
    const int j = t & 3;
    const int m = (t >> 2) & 63;
    const int pc = t >> 8;  // pos*(CIN/4) + cb
    const int cb = pc % (CIN / 4);
    const int pos = pc / (CIN / 4);
    const int cin = cb * 4 + j;
    const int ky = pos / 3, kx = pos % 3;
    dst[t] = src[((m * CIN + cin) * 3 + ky) * 3 + kx];
}

// ---------------- implicit-GEMM 3x3 conv on V_WMMA_F32_16X16X4_F32 ----------
// Block = 128 threads (4 wave32). Tile = 64 couts x 16 output pixels (one row).
// Wave wv computes couts [16*wv, 16*wv+16). K = 9*CIN, stepped by 4.
// A-frag (16x4 f32): lane<16 holds (m=lane, k0+0..1); lane>=16 (m=lane-16, k0+2..3).
// B-frag (4x16 f32): mirrored K split, N = lane&15.
// C/D (16x16 f32): VGPR r -> (M = r + 8*(lane>=16), N = lane&15).
// Interior tiles stage the input halo via GLOBAL_LOAD_ASYNC_TO_LDS_B32
// (no VGPR round-trip, tracked by ASYNCcnt); boundary tiles use a guarded
// scalar path that synthesizes the zero padding.
template <int CIN, int STRIDE, bool UPCAT>
__global__ __launch_bounds__(128) void wmma_conv3x3(
    const float* __restrict__ in0,   // plain input, or (UPCAT) half-res normalized bm
    const float* __restrict__ in1,   // (UPCAT) full-res c1n for channels 64..95
    const float* __restrict__ wtB,   // swizzled weights [pos][CIN/4][64][4]
    const float* __restrict__ bias,  // 64
    float* __restrict__ out,         // 64 x outH x outW, raw conv+bias
    int inH, int inW, int outH, int outW) {
    constexpr int TW = (STRIDE == 1) ? 18 : 34;  // tile width incl. halo
    constexpr int CSTR = 3 * TW;                 // floats per input channel in LDS
    __shared__ float tile[CIN * CSTR];

    const int tid = threadIdx.x;
    const int x0 = blockIdx.x * 16;
    const int y = blockIdx.y;
    constexpr int pbeg = (STRIDE == 1) ? 1 : 0;  // jax 'SAME' padding start
    const int rowbase = y * STRIDE - pbeg;
    const int colbase = x0 * STRIDE - pbeg;

    // ---- stage input halo tile into LDS ----
    const bool interior = (rowbase >= 0) && (rowbase + 3 <= inH) &&
                          (colbase >= 0) && (colbase + TW <= inW);
    if (interior) {
        // async direct-to-LDS staging (no VGPR return, ASYNCcnt-tracked)
        for (int i = tid; i < CIN * CSTR; i += 128) {
            const int cin = i / CSTR;
            const int rem = i - cin * CSTR;
            const int rr = rem / TW;
            const int cc = rem - rr * TW;
            const int gy = rowbase + rr;
            const int gx = colbase + cc;
            const float* gp;
            if (UPCAT) {
                gp = (cin < 64)
                         ? in0 + (size_t)cin * ((inH / 2) * (inW / 2)) + (gy >> 1) * (inW / 2) + (gx >> 1)
                         : in1 + (size_t)(cin - 64) * (inH * inW) + (size_t)gy * inW + gx;
            } else {
                gp = in0 + (size_t)cin * (inH * inW) + (size_t)gy * inW + gx;
            }
            const unsigned lds_off = (unsigned)(uintptr_t)&tile[i];  // low 32b of flat = LDS offset
            asm volatile("global_load_async_to_lds_b32 %0, %1, off"
                         :: "v"(lds_off), "v"(gp)
                         : "memory");
        }
        asm volatile("s_wait_asynccnt 0x0" ::: "memory");
    } else {
        for (int i = tid; i < CIN * CSTR; i += 128) {
            const int cin = i / CSTR;
            const int rem = i - cin * CSTR;
            const int rr = rem / TW;
            const int cc = rem - rr * TW;
            const int gy = rowbase + rr;
            const int gx = colbase + cc;
            float v = 0.f;
            if ((unsigned)gy < (unsigned)inH && (unsigned)gx < (unsigned)inW) {
                if (UPCAT) {
                    if (cin < 64)  // nearest-upsampled half-res feature map
                        v = in0[(size_t)cin * ((inH / 2) * (inW / 2)) + (gy >> 1) * (inW / 2) + (gx >> 1)];
                    else
                        v = in1[(size_t)(cin - 64) * (inH * inW) + (size_t)gy * inW + gx];
                } else {
                    v = in0[(size_t)cin * (inH * inW) + (size_t)gy * inW + gx];
                }
            }
            tile[i] = v;
        }
    }
    __syncthreads();

    const int lane = tid & 31;
    const int wv = tid >> 5;
    const int half = lane >> 4;
    const int n = lane & 15;
    const int koff = half * 2;          // K offset of this half-wave within the 4-step
    const int m = wv * 16 + n;          // A-matrix row (cout)
    const int cstart = STRIDE * n;      // pixel column base inside LDS tile

    v8f acc = {};
    const float* wp = wtB + m * 4 + koff;

#pragma unroll
    for (int pos = 0; pos < 9; ++pos) {
        const int posoff = (pos / 3) * TW + (pos % 3);
        const float* bpt = &tile[koff * CSTR + posoff + cstart];
#pragma unroll 4
        for (int cb = 0; cb < CIN / 4; ++cb) {
            v2f a = *(const v2f*)(wp + (size_t)(pos * (CIN / 4) + cb) * 256);
            v2f b;
            b.x = bpt[cb * 4 * CSTR];
            b.y = bpt[cb * 4 * CSTR + CSTR];
            acc = __builtin_amdgcn_wmma_f32_16x16x4_f32(
                /*neg_a=*/false, a, /*neg_b=*/false, b,
                /*c_mod=*/(short)0, acc, /*reuse_a=*/false, /*reuse_b=*/false);
        }
    }

    const int x = x0 + n;
    if (x < outW) {
#pragma unroll
        for (int r = 0; r < 8; ++r) {
            const int cout = wv * 16 + half * 8 + r;
            out[(size_t)cout * (outH * outW) + (size_t)y * outW + x] = acc[r] + bias[cout];
        }
    }
}

// ---------------- per-pair head --------------------------------------------

__global__ __launch_bounds__(256) void head_kernel(
    const float* __restrict__ pose, const int* __restrict__ pairs,
    const float* __restrict__ depth, const float* __restrict__ meanP,
    const float* __restrict__ feats, const float* __restrict__ aff4,
    const float* __restrict__ gw, const float* __restrict__ gb,
    const float* __restrict__ hw, const float* __restrict__ hb,
    const float* __restrict__ ow, const float* __restrict__ ob,
    float* __restrict__ outp, int P, int S) {
    __shared__ float sgw[15 * 64], shw[64 * 64], sgb[64], shb[64], sow[64];
    for (int i = threadIdx.x; i < 15 * 64; i += 256) sgw[i] = gw[i];
    for (int i = threadIdx.x; i < 64 * 64; i += 256) shw[i] = hw[i];
    if (threadIdx.x < 64) {
        sgb[threadIdx.x] = gb[threadIdx.x];
        shb[threadIdx.x] = hb[threadIdx.x];
        sow[threadIdx.x] = ow[threadIdx.x];
    }
    __syncthreads();

    const int r = blockIdx.x * 256 + threadIdx.x;
    if (r >= 2 * P) return;
    const int p = r >> 1;
    const int idx = pairs[2 * p];
    const float* ps = pose + (size_t)r * 8;

    const float q0 = ps[0], q1 = ps[1], q2 = ps[2], q3 = ps[3];
    float cond[15];
    cond[0] = q0 * q0; cond[1] = q0 * q1; cond[2] = q0 * q2; cond[3] = q0 * q3;
    cond[4] = q1 * q1; cond[5] = q1 * q2; cond[6] = q1 * q3;
    cond[7] = q2 * q2; cond[8] = q2 * q3; cond[9] = q3 * q3;
    cond[10] = ps[4]; cond[11] = ps[5]; cond[12] = ps[6]; cond[13] = ps[7];
    cond[14] = (depth[idx] - *meanP) * 10.f;

    float h[64];
    float ss = 0.f;
#pragma unroll
    for (int c = 0; c < 64; ++c) {
        float g = sgb[c];
#pragma unroll
        for (int i = 0; i < 15; ++i) g += cond[i] * sgw[i * 64 + c];
        g = 1.f / (1.f + __expf(-g));
        const float A = aff4[2 * c], B = aff4[2 * c + 1];
        float t = feats[(size_t)c * S + idx] * A + B;
        t = LRELU(t);
        const float hv = t * g;
        h[c] = hv;
        ss += hv * hv;
    }
    const float rn = rsqrtf(ss + 1e-8f);
#pragma unroll
    for (int c = 0; c < 64; ++c) h[c] *= rn;

    float res = *ob;
    for (int c2 = 0; c2 < 64; ++c2) {
        float t = shb[c2];
#pragma unroll
        for (int k = 0; k < 64; ++k) t += h[k] * shw[k * 64 + c2];
        t = LRELU(t);
        res += (t / (1.f + __expf(-t))) * sow[c2];  // silu(t) * ow
    }
    outp[r] = res;
}

// ---------------------------------------------------------------------------

extern "C" void kernel_launch(void* const* d_in, const int* in_sizes, int n_in,
                              void* d_out, int out_size, void* d_ws, size_t ws_size,
                              hipStream_t stream) {
    const int H = 480, W = 600, P = 2048;
    const int S0 = H * W;            // 288000
    const int H1 = 240, W1 = 300;
    const int S1 = H1 * W1;          // 72000

    const float* depth = (const float*)d_in[0];
    const float* pose  = (const float*)d_in[1];
    const float* mask  = (const float*)d_in[2];
    const int*   pairs = (const int*)d_in[3];
    const float* c1w = (const float*)d_in[4];  const float* c1b = (const float*)d_in[5];
    const float* g1s = (const float*)d_in[6];  const float* g1b = (const float*)d_in[7];
    const float* c2w = (const float*)d_in[8];  const float* c2b = (const float*)d_in[9];
    const float* g2s = (const float*)d_in[10]; const float* g2b = (const float*)d_in[11];
    const float* c3w = (const float*)d_in[12]; const float* c3b = (const float*)d_in[13];
    const float* g3s = (const float*)d_in[14]; const float* g3b = (const float*)d_in[15];
    const float* c4w = (const float*)d_in[16]; const float* c4b = (const float*)d_in[17];
    const float* g4s = (const float*)d_in[18]; const float* g4b = (const float*)d_in[19];
    const float* gw  = (const float*)d_in[20]; const float* gb  = (const float*)d_in[21];
    const float* hw_ = (const float*)d_in[22]; const float* hb  = (const float*)d_in[23];
    const float* ow  = (const float*)d_in[24]; const float* ob  = (const float*)d_in[25];

    float* ws = (float*)d_ws;

    // ---- stats region (float indices) ----
    float* meanP = ws;                 // [0]
    float* AFF1  = ws + 16;            // 32 x float2
    float* AFF2  = ws + 80;            // 64 x float2
    float* AFF3  = ws + 208;
    float* AFF4  = ws + 336;
    float* PSUM  = ws + 1024;          // 64 blk x 64 ch partial sums
    float* PSQ   = ws + 5120;          // partial sum-of-squares
    // ---- tensor regions (overlapped by liveness) ----
    const size_t T0 = 10240;                       // x (3xS0) then feats (64xS0)
    const size_t T1 = T0 + (size_t)64 * S0;        // c1raw (32xS0) then d1raw (64xS1)
    const size_t T2 = T1 + (size_t)32 * S0;        // c1n (32xS0)
    const size_t T3 = T2 + (size_t)32 * S0;        // d1n then bmn (64xS1)
    const size_t T4 = T3 + (size_t)64 * S1;        // bmraw (64xS1)
    const size_t T5 = T4 + (size_t)64 * S1;        // wtB2 (9*8*256)
    const size_t T6 = T5 + 18432;                  // wtB3 (9*16*256)
    const size_t T7 = T6 + 36864;                  // wtB4 (9*24*256)
    float* xbuf  = ws + T0;  float* feats = ws + T0;
    float* c1raw = ws + T1;  float* d1raw = ws + T1;
    float* c1n   = ws + T2;
    float* d1n   = ws + T3;  float* bmn   = ws + T3;
    float* bmraw = ws + T4;
    float* wtB2  = ws + T5;
    float* wtB3  = ws + T6;
    float* wtB4  = ws + T7;

    const int B0 = (S0 + 255) / 256;   // 1125
    const int B1 = (S1 + 255) / 256;   // 282

    // depth mean (deterministic two-stage)
    depth_partials<<<256, 256, 0, stream>>>(depth, S0, PSUM);
    finalize_mean<<<1, 32, 0, stream>>>(PSUM, 256, 1.f / (float)S0, meanP);

    // network input: [std_depth, mask, attn]
    build_x<<<B0, 256, 0, stream>>>(depth, mask, meanP, xbuf, S0);
    scatter_attn<<<(P + 255) / 256, 256, 0, stream>>>(pairs, xbuf + 2 * (size_t)S0, P);

    // weight swizzles for the WMMA convs
    prep_weights<<<(64 * 9 * 32 + 255) / 256, 256, 0, stream>>>(c2w, wtB2, 32);
    prep_weights<<<(64 * 9 * 64 + 255) / 256, 256, 0, stream>>>(c3w, wtB3, 64);
    prep_weights<<<(64 * 9 * 96 + 255) / 256, 256, 0, stream>>>(c4w, wtB4, 96);

    // conv1 + GN1 + lrelu
    conv1_direct<<<dim3(B0, 32), 256, 0, stream>>>(xbuf, c1w, c1b, c1raw, H, W);
    chan_partials<<<dim3(64, 32), 256, 0, stream>>>(c1raw, S0, PSUM, PSQ);
    finalize_gn<<<1, 32, 0, stream>>>(PSUM, PSQ, 64, 32, 2, (float)S0, g1s, g1b, AFF1);
    apply_gn_lrelu<<<dim3(B0, 32), 256, 0, stream>>>(c1raw, AFF1, c1n, S0);

    // conv2 (stride 2, WMMA) + GN2 + lrelu
    wmma_conv3x3<32, 2, false><<<dim3((W1 + 15) / 16, H1), 128, 0, stream>>>(
        c1n, nullptr, wtB2, c2b, d1raw, H, W, H1, W1);
    chan_partials<<<dim3(64, 64), 256, 0, stream>>>(d1raw, S1, PSUM, PSQ);
    finalize_gn<<<1, 64, 0, stream>>>(PSUM, PSQ, 64, 64, 4, (float)S1, g2s, g2b, AFF2);
    apply_gn_lrelu<<<dim3(B1, 64), 256, 0, stream>>>(d1raw, AFF2, d1n, S1);

    // conv3 (WMMA) + GN3 + lrelu
    wmma_conv3x3<64, 1, false><<<dim3((W1 + 15) / 16, H1), 128, 0, stream>>>(
        d1n, nullptr, wtB3, c3b, bmraw, H1, W1, H1, W1);
    chan_partials<<<dim3(64, 64), 256, 0, stream>>>(bmraw, S1, PSUM, PSQ);
    finalize_gn<<<1, 64, 0, stream>>>(PSUM, PSQ, 64, 64, 4, (float)S1, g3s, g3b, AFF3);
    apply_gn_lrelu<<<dim3(B1, 64), 256, 0, stream>>>(bmraw, AFF3, bmn, S1);

    // conv4 (WMMA, fused nearest-upsample + concat loader) -> raw feats
    wmma_conv3x3<96, 1, true><<<dim3((W + 15) / 16, H), 128, 0, stream>>>(
        bmn, c1n, wtB4, c4b, feats, H, W, H, W);
    chan_partials<<<dim3(64, 64), 256, 0, stream>>>(feats, S0, PSUM, PSQ);
    finalize_gn<<<1, 64, 0, stream>>>(PSUM, PSQ, 64, 64, 4, (float)S0, g4s, g4b, AFF4);

    // per-pair head (GN4+lrelu applied at gather time)
    head_kernel<<<(2 * P + 255) / 256, 256, 0, stream>>>(
        pose, pairs, depth, meanP, feats, AFF4, gw, gb, hw_, hb, ow, ob,
        (float*)d_out, P, S0);
}